// GlobalLocalCrossAttention_26431228740296
// MI455X (gfx1250) — compile-verified
//
#include <hip/hip_runtime.h>
#include <cstddef>
#include <cstdint>

// Problem constants (match reference)
#define B_    16
#define N_    785
#define C_    768
#define H_    12
#define D_    64
#define S_    79      // 1 CLS + 78 selected patches
#define NP_   784
#define NSEL_ 78
#define SCALE_ 0.125f // 64^-0.5

typedef float v2f __attribute__((ext_vector_type(2)));
typedef float v8f __attribute__((ext_vector_type(8)));

// ---------------------------------------------------------------------------
// CDNA5 async global->LDS helpers (ASYNCcnt-tracked, no VGPR staging).
// Generic LDS address low 32 bits == LDS offset (ISA aperture mapping).
// ---------------------------------------------------------------------------
__device__ __forceinline__ uint32_t lds_off_of(const void* p) {
  return (uint32_t)(uintptr_t)p;
}
__device__ __forceinline__ void async_g2l_b64(uint32_t lds_byte_off, const float* g) {
  asm volatile("global_load_async_to_lds_b64 %0, %1, off"
               :: "v"(lds_byte_off), "v"((uint64_t)(uintptr_t)g)
               : "memory");
}
__device__ __forceinline__ void wait_async_le8() {
  asm volatile("s_wait_asynccnt 0x8" ::: "memory");
}
__device__ __forceinline__ void wait_async_0() {
  asm volatile("s_wait_asynccnt 0x0" ::: "memory");
}

// ---------------------------------------------------------------------------
// 1) Rollout stage A: m1[b,i,j] = mean_h attn[0,b,h,i,j]; r1[b,i] = 0.5*rowsum+0.5
//    One block per (b,i) row; streams 473 MB of layer-0 history (the BW floor).
// ---------------------------------------------------------------------------
__global__ __launch_bounds__(256) void rollout_mean_rowsum(
    const float* __restrict__ attn0, float* __restrict__ m1, float* __restrict__ r1) {
  const int b = blockIdx.x / N_;
  const int i = blockIdx.x % N_;
  const int tid = threadIdx.x;
  const size_t rowbase = ((size_t)(b * H_) * N_ + i) * (size_t)N_;
  float part = 0.0f;
  for (int j = tid; j < N_; j += 256) {
    float s = 0.0f;
#pragma unroll
    for (int h = 0; h < H_; ++h)
      s += attn0[rowbase + (size_t)h * N_ * N_ + j];
    s *= (1.0f / (float)H_);
    m1[((size_t)b * N_ + i) * N_ + j] = s;
    part += s;
  }
  __shared__ float red[256];
  red[tid] = part; __syncthreads();
  for (int s2 = 128; s2 > 0; s2 >>= 1) {
    if (tid < s2) red[tid] += red[tid + s2];
    __syncthreads();
  }
  if (tid == 0) r1[b * N_ + i] = 0.5f * red[0] + 0.5f;
}

// ---------------------------------------------------------------------------
// 2) w[b,i] = a2_row0[i] / r1[b,i]
// ---------------------------------------------------------------------------
__global__ __launch_bounds__(256) void rollout_w(
    const float* __restrict__ attn1, const float* __restrict__ r1, float* __restrict__ w) {
  const int b = blockIdx.x;
  const int tid = threadIdx.x;
  __shared__ float m2[N_];
  __shared__ float red[256];
  float part = 0.0f;
  for (int i = tid; i < N_; i += 256) {
    float s = 0.0f;
#pragma unroll
    for (int h = 0; h < H_; ++h)
      s += attn1[(((size_t)b * H_ + h) * N_ + 0) * N_ + i];
    s *= (1.0f / (float)H_);
    m2[i] = s;
    part += s;
  }
  red[tid] = part; __syncthreads();
  for (int s2 = 128; s2 > 0; s2 >>= 1) {
    if (tid < s2) red[tid] += red[tid + s2];
    __syncthreads();
  }
  const float r2inv = 1.0f / (0.5f * red[0] + 0.5f);
  for (int i = tid; i < N_; i += 256) {
    float a2 = (0.5f * m2[i] + (i == 0 ? 0.5f : 0.0f)) * r2inv;
    w[b * N_ + i] = a2 / r1[b * N_ + i];
  }
}

// ---------------------------------------------------------------------------
// 3) scores[b,j] = 0.5 * sum_i w[b,i]*m1[b,i,j] + 0.5*w[b,j]
// ---------------------------------------------------------------------------
__global__ __launch_bounds__(256) void rollout_scores(
    const float* __restrict__ m1, const float* __restrict__ w, float* __restrict__ scores) {
  const int b = blockIdx.y;
  const int j = blockIdx.x * 256 + threadIdx.x;
  if (j >= N_) return;
  const float* wb = w + b * N_;
  const float* mb = m1 + (size_t)b * N_ * N_ + j;
  float acc = 0.0f;
  for (int i = 0; i < N_; ++i) {
    if ((i & 63) == 0 && i + 64 < N_) __builtin_prefetch(mb + (size_t)(i + 64) * N_, 0, 0);
    acc += wb[i] * mb[(size_t)i * N_];
  }
  scores[b * N_ + j] = 0.5f * acc + 0.5f * wb[j];
}

// ---------------------------------------------------------------------------
// 4) Top-78 per batch (descending, lowest-index tie-break like jax.lax.top_k)
// ---------------------------------------------------------------------------
__global__ __launch_bounds__(256) void topk78(
    const float* __restrict__ scores, int* __restrict__ sel) {
  const int b = blockIdx.x;
  const int tid = threadIdx.x;
  __shared__ float vals[NP_];
  __shared__ float rv[256];
  __shared__ int   ri[256];
  for (int j = tid; j < NP_; j += 256) vals[j] = scores[b * N_ + 1 + j];
  if (tid == 0) sel[b * S_] = 0;
  __syncthreads();
  for (int t = 0; t < NSEL_; ++t) {
    float best = -3.4e38f; int bi = NP_;
    for (int j = tid; j < NP_; j += 256) {
      float v = vals[j];
      if (v > best || (v == best && j < bi)) { best = v; bi = j; }
    }
    rv[tid] = best; ri[tid] = bi; __syncthreads();
    for (int s2 = 128; s2 > 0; s2 >>= 1) {
      if (tid < s2) {
        if (rv[tid + s2] > rv[tid] || (rv[tid + s2] == rv[tid] && ri[tid + s2] < ri[tid])) {
          rv[tid] = rv[tid + s2]; ri[tid] = ri[tid + s2];
        }
      }
      __syncthreads();
    }
    if (tid == 0) { sel[b * S_ + 1 + t] = ri[0] + 1; vals[ri[0]] = -3.4e38f; }
    __syncthreads();
  }
}

// ---------------------------------------------------------------------------
// 5) Gather selected query rows
// ---------------------------------------------------------------------------
__global__ __launch_bounds__(256) void gather_rows(
    const float* __restrict__ x, const int* __restrict__ sel, float* __restrict__ qg) {
  const int r = blockIdx.x;
  const int b = r / S_, s = r % S_;
  const int tok = sel[b * S_ + s];
  const float* src = x + ((size_t)b * N_ + tok) * C_;
  float* dst = qg + (size_t)r * C_;
  for (int c = threadIdx.x; c < C_; c += 256) dst[c] = src[c];
}

// ---------------------------------------------------------------------------
// 6) WMMA fp32 GEMM with double-buffered ASYNC global->LDS tiles.
//    C[M,N] = A[M,K] @ W^T + bias  (W is [N,K] row-major; N,K multiples of 128/16)
//    Block tile 128x128, 8 waves; wave = 16(M) x 128(N) strip = 8 wmma tiles.
//    LDS rows padded to 18 floats; OOB A rows are address-clamped (outputs of
//    those rows are masked at the store, so garbage never escapes).
// ---------------------------------------------------------------------------
#define BM_   128
#define BN_   128
#define BK_   16
#define LDST_ 18

__device__ __forceinline__ void issue_tile_async(
    const float* A, const float* W, int M, int K, int m0, int n0, int k0,
    uint32_t asBase, uint32_t bsBase, int tid) {
#pragma unroll
  for (int i = 0; i < 4; ++i) {          // A tile: 128 x 16 floats as 1024 B64 ops
    const int c = tid + i * 256;         // 0..1023
    const int m = c >> 3, kq = (c & 7) * 2;
    int gr = m0 + m; if (gr >= M) gr = M - 1;  // clamp (safe, see note)
    async_g2l_b64(asBase + (uint32_t)(m * LDST_ + kq) * 4,
                  A + (size_t)gr * K + k0 + kq);
  }
#pragma unroll
  for (int i = 0; i < 4; ++i) {          // W tile: 128 x 16 floats
    const int c = tid + i * 256;
    const int n = c >> 3, kq = (c & 7) * 2;
    async_g2l_b64(bsBase + (uint32_t)(n * LDST_ + kq) * 4,
                  W + (size_t)(n0 + n) * K + k0 + kq);
  }
}

__global__ __launch_bounds__(256) void gemm_xwT_wmma(
    const float* __restrict__ A, const float* __restrict__ W,
    const float* __restrict__ bias, float* __restrict__ C, int M, int N, int K) {
  __shared__ float smem[2 * (BM_ + BN_) * LDST_];   // 36,864 B
  const uint32_t base = lds_off_of(smem);
  const uint32_t aBytes = BM_ * LDST_ * 4;
  const uint32_t bufBytes = (BM_ + BN_) * LDST_ * 4;
  const int tid = threadIdx.x;
  const int lane = tid & 31;
  const int wave = tid >> 5;
  const int m0 = blockIdx.x * BM_;
  const int n0 = blockIdx.y * BN_;

  v8f acc[8] = {};

  issue_tile_async(A, W, M, K, m0, n0, 0, base, base + aBytes, tid);

  const int nchunks = K / BK_;
  for (int ch = 0; ch < nchunks; ++ch) {
    if (ch + 1 < nchunks) {
      const uint32_t nb = base + (uint32_t)((ch + 1) & 1) * bufBytes;
      issue_tile_async(A, W, M, K, m0, n0, (ch + 1) * BK_, nb, nb + aBytes, tid);
      wait_async_le8();   // in-order completion => current chunk's 8 ops retired
    } else {
      wait_async_0();
    }
    __syncthreads();

    const float* As = smem + (size_t)(ch & 1) * (BM_ + BN_) * LDST_;
    const float* Bs = As + BM_ * LDST_;
    const int mrow  = wave * 16 + (lane & 15);
    const int khalf = (lane >> 4) * 2;   // lanes 0-15: K{0,1}; 16-31: K{2,3}
#pragma unroll
    for (int kk = 0; kk < 4; ++kk) {
      v2f a = *(const v2f*)&As[mrow * LDST_ + kk * 4 + khalf];
#pragma unroll
      for (int nt = 0; nt < 8; ++nt) {
        v2f bfr = *(const v2f*)&Bs[(nt * 16 + (lane & 15)) * LDST_ + kk * 4 + khalf];
        acc[nt] = __builtin_amdgcn_wmma_f32_16x16x4_f32(
            false, a, false, bfr, (short)0, acc[nt], false, false);
      }
    }
    __syncthreads();   // protect buffer from next-next issue
  }

  const int colb = n0 + (lane & 15);
  const int rowb = m0 + wave * 16 + ((lane >> 4) << 3);
#pragma unroll
  for (int nt = 0; nt < 8; ++nt) {
    const int col = colb + nt * 16;
    const float bv = bias ? bias[col] : 0.0f;
#pragma unroll
    for (int r = 0; r < 8; ++r) {
      const int row = rowb + r;
      if (row < M) C[(size_t)row * N + col] = acc[nt][r] + bv;
    }
  }
}

// ---------------------------------------------------------------------------
// 7) Attention logits: logits[b,h,s,n] = scale * q.k  (into d_out slice)
// ---------------------------------------------------------------------------
__global__ __launch_bounds__(256) void attn_logits_wmma(
    const float* __restrict__ Q, const float* __restrict__ Kb, float* __restrict__ logits) {
  __shared__ float Qs[16][18];
  __shared__ float Ks[128][18];
  const int tid = threadIdx.x;
  const int lane = tid & 31;
  const int wave = tid >> 5;
  const int n0 = blockIdx.x * 128;
  const int s0 = blockIdx.y * 16;
  const int bh = blockIdx.z;
  const int b = bh / H_, h = bh % H_;

  v8f acc = {};

  for (int k0 = 0; k0 < D_; k0 += 16) {
    {
      const int s = tid >> 4, k = tid & 15;
      Qs[s][k] = (s0 + s < S_) ? Q[((size_t)b * S_ + s0 + s) * C_ + h * D_ + k0 + k] : 0.0f;
    }
    {
      const int n = tid >> 1, kp = (tid & 1) * 8;
      float* dst = &Ks[n][kp];
      if (n0 + n < N_) {
        const float* src = Kb + ((size_t)b * N_ + n0 + n) * C_ + h * D_ + k0 + kp;
#pragma unroll
        for (int i = 0; i < 8; ++i) dst[i] = src[i];
      } else {
#pragma unroll
        for (int i = 0; i < 8; ++i) dst[i] = 0.0f;
      }
    }
    __syncthreads();

    const int khalf = (lane >> 4) * 2;
#pragma unroll
    for (int kk = 0; kk < 4; ++kk) {
      v2f a = *(const v2f*)&Qs[lane & 15][kk * 4 + khalf];
      v2f bfr = *(const v2f*)&Ks[wave * 16 + (lane & 15)][kk * 4 + khalf];
      acc = __builtin_amdgcn_wmma_f32_16x16x4_f32(false, a, false, bfr, (short)0, acc, false, false);
    }
    __syncthreads();
  }

  const int n = n0 + wave * 16 + (lane & 15);
  const int sb = s0 + ((lane >> 4) << 3);
#pragma unroll
  for (int r = 0; r < 8; ++r) {
    const int s = sb + r;
    if (s < S_ && n < N_)
      logits[((size_t)bh * S_ + s) * N_ + n] = acc[r] * SCALE_;
  }
}

// ---------------------------------------------------------------------------
// 8) Row softmax in place
// ---------------------------------------------------------------------------
__global__ __launch_bounds__(256) void softmax_rows(float* __restrict__ p_all, int n) {
  float* p = p_all + (size_t)blockIdx.x * n;
  const int tid = threadIdx.x;
  __shared__ float red[256];
  float m = -3.4e38f;
  for (int j = tid; j < n; j += 256) m = fmaxf(m, p[j]);
  red[tid] = m; __syncthreads();
  for (int s2 = 128; s2 > 0; s2 >>= 1) {
    if (tid < s2) red[tid] = fmaxf(red[tid], red[tid + s2]);
    __syncthreads();
  }
  const float mx = red[0]; __syncthreads();
  float sum = 0.0f;
  for (int j = tid; j < n; j += 256) { float e = __expf(p[j] - mx); p[j] = e; sum += e; }
  red[tid] = sum; __syncthreads();
  for (int s2 = 128; s2 > 0; s2 >>= 1) {
    if (tid < s2) red[tid] += red[tid + s2];
    __syncthreads();
  }
  const float inv = 1.0f / red[0];
  for (int j = tid; j < n; j += 256) p[j] *= inv;
}

// ---------------------------------------------------------------------------
// 9) out_heads = attn_weights @ V  (K-dim tail must be zero-padded => sync path)
// ---------------------------------------------------------------------------
__global__ __launch_bounds__(128) void attn_av_wmma(
    const float* __restrict__ AW, const float* __restrict__ V, float* __restrict__ O) {
  __shared__ float As[16][18];
  __shared__ float Bs[64][18];
  const int tid = threadIdx.x;
  const int lane = tid & 31;
  const int wave = tid >> 5;
  const int s0 = blockIdx.x * 16;
  const int bh = blockIdx.y;
  const int b = bh / H_, h = bh % H_;

  v8f acc = {};

  for (int k0 = 0; k0 < N_; k0 += 16) {
    {
      const int idx = tid * 2;
      const int s = idx >> 4, k = idx & 15;
#pragma unroll
      for (int i = 0; i < 2; ++i) {
        const int kk = k + i;
        As[s][kk] = (s0 + s < S_ && k0 + kk < N_)
                        ? AW[((size_t)bh * S_ + s0 + s) * N_ + k0 + kk] : 0.0f;
      }
    }
    {
      const int k = tid >> 3, dq = (tid & 7) * 8;
      const bool ok = (k0 + k) < N_;
      const float* src = V + ((size_t)b * N_ + k0 + k) * C_ + h * D_ + dq;
#pragma unroll
      for (int i = 0; i < 8; ++i) Bs[dq + i][k] = ok ? src[i] : 0.0f;
    }
    __syncthreads();

    const int khalf = (lane >> 4) * 2;
#pragma unroll
    for (int kk = 0; kk < 4; ++kk) {
      v2f a = *(const v2f*)&As[lane & 15][kk * 4 + khalf];
      v2f bfr = *(const v2f*)&Bs[wave * 16 + (lane & 15)][kk * 4 + khalf];
      acc = __builtin_amdgcn_wmma_f32_16x16x4_f32(false, a, false, bfr, (short)0, acc, false, false);
    }
    __syncthreads();
  }

  const int d = wave * 16 + (lane & 15);
  const int sb = s0 + ((lane >> 4) << 3);
#pragma unroll
  for (int r = 0; r < 8; ++r) {
    const int s = sb + r;
    if (s < S_)
      O[((size_t)b * S_ + s) * C_ + h * D_ + d] = acc[r];
  }
}

// ---------------------------------------------------------------------------
// 10) zero + scatter
// ---------------------------------------------------------------------------
__global__ __launch_bounds__(256) void zero_f32(float* __restrict__ p, long long n) {
  long long i = (long long)blockIdx.x * 256 + threadIdx.x;
  const long long stride = (long long)gridDim.x * 256;
  for (; i < n; i += stride) p[i] = 0.0f;
}

__global__ __launch_bounds__(256) void scatter_rows(
    const float* __restrict__ fbuf, const int* __restrict__ sel, float* __restrict__ out) {
  const int r = blockIdx.x;
  const int b = r / S_, s = r % S_;
  const int tok = sel[b * S_ + s];
  const float* src = fbuf + (size_t)r * C_;
  float* dst = out + ((size_t)b * N_ + tok) * C_;
  for (int c = threadIdx.x; c < C_; c += 256) dst[c] = src[c];
}

// ---------------------------------------------------------------------------
extern "C" void kernel_launch(void* const* d_in, const int* in_sizes, int n_in,
                              void* d_out, int out_size, void* d_ws, size_t ws_size,
                              hipStream_t stream) {
  (void)in_sizes; (void)n_in; (void)out_size; (void)ws_size;
  const float* x   = (const float*)d_in[0];
  const float* ah  = (const float*)d_in[1];
  const float* Wq  = (const float*)d_in[2];
  const float* bq  = (const float*)d_in[3];
  const float* Wk  = (const float*)d_in[4];
  const float* bk  = (const float*)d_in[5];
  const float* Wv  = (const float*)d_in[6];
  const float* bv  = (const float*)d_in[7];
  const float* Wo  = (const float*)d_in[8];
  const float* bo  = (const float*)d_in[9];

  float* out   = (float*)d_out;                       // [B,N,C]
  float* attnw = out + (size_t)B_ * N_ * C_;          // [B,H,S,N]

  float* ws = (float*)d_ws;
  size_t off = 0;
  float* m1     = ws + off; off += (size_t)B_ * N_ * N_;
  float* r1     = ws + off; off += (size_t)B_ * N_;
  float* wvec   = ws + off; off += (size_t)B_ * N_;
  float* scores = ws + off; off += (size_t)B_ * N_;
  int*   sel    = (int*)(ws + off); off += (size_t)B_ * S_;
  float* kbuf   = ws + off; off += (size_t)B_ * N_ * C_;
  float* vbuf   = ws + off; off += (size_t)B_ * N_ * C_;
  float* qg     = ws + off; off += (size_t)B_ * S_ * C_;
  float* qbuf   = ws + off; off += (size_t)B_ * S_ * C_;
  float* aout   = ws + off; off += (size_t)B_ * S_ * C_;
  float* fbuf   = ws + off; off += (size_t)B_ * S_ * C_;

  const float* attn0 = ah;
  const float* attn1 = ah + (size_t)B_ * H_ * N_ * N_;

  // ---- rollout (bandwidth stage) ----
  rollout_mean_rowsum<<<B_ * N_, 256, 0, stream>>>(attn0, m1, r1);
  rollout_w<<<B_, 256, 0, stream>>>(attn1, r1, wvec);
  rollout_scores<<<dim3((N_ + 255) / 256, B_), 256, 0, stream>>>(m1, wvec, scores);
  topk78<<<B_, 256, 0, stream>>>(scores, sel);

  // ---- projections (async-tiled WMMA fp32) ----
  const int MKV = B_ * N_;   // 12560
  const int MQ  = B_ * S_;   // 1264
  gemm_xwT_wmma<<<dim3((MKV + BM_ - 1) / BM_, C_ / BN_), 256, 0, stream>>>(x, Wk, bk, kbuf, MKV, C_, C_);
  gemm_xwT_wmma<<<dim3((MKV + BM_ - 1) / BM_, C_ / BN_), 256, 0, stream>>>(x, Wv, bv, vbuf, MKV, C_, C_);
  gather_rows<<<MQ, 256, 0, stream>>>(x, sel, qg);
  gemm_xwT_wmma<<<dim3((MQ + BM_ - 1) / BM_, C_ / BN_), 256, 0, stream>>>(qg, Wq, bq, qbuf, MQ, C_, C_);

  // ---- attention (WMMA) + softmax ----
  attn_logits_wmma<<<dim3((N_ + 127) / 128, (S_ + 15) / 16, B_ * H_), 256, 0, stream>>>(qbuf, kbuf, attnw);
  softmax_rows<<<B_ * H_ * S_, 256, 0, stream>>>(attnw, N_);
  attn_av_wmma<<<dim3((S_ + 15) / 16, B_ * H_), 128, 0, stream>>>(attnw, vbuf, aout);

  // ---- output projection + scatter ----
  gemm_xwT_wmma<<<dim3((MQ + BM_ - 1) / BM_, C_ / BN_), 256, 0, stream>>>(aout, Wo, bo, fbuf, MQ, C_, C_);
  zero_f32<<<4096, 256, 0, stream>>>(out, (long long)B_ * N_ * C_);
  scatter_rows<<<MQ, 256, 0, stream>>>(fbuf, sel, out);
}